// MultiScaleGNNBlock_29308856828502
// MI455X (gfx1250) — compile-verified
//
#include <hip/hip_runtime.h>
#include <math.h>

typedef __attribute__((ext_vector_type(2))) float v2f;
typedef __attribute__((ext_vector_type(8))) float v8f;

// fp32 WMMA: D(16x16,f32) = A(16x4,f32) x B(4x16,f32) + C
// 8-arg pattern per probe doc: (neg_a, A, neg_b, B, c_mod, C, reuse_a, reuse_b)
#define WMMA4(a, b, c) \
  __builtin_amdgcn_wmma_f32_16x16x4_f32(false, (a), false, (b), (short)0, (c), false, false)

__device__ __forceinline__ float gelu_exact(float v) {
  return 0.5f * v * (1.0f + erff(v * 0.70710678118654752440f));
}

// ---------------------------------------------------------------- utilities
__global__ void k_zero(float* __restrict__ p, long n) {
  long i = (long)blockIdx.x * blockDim.x + threadIdx.x;
  if (i < n) p[i] = 0.0f;
}

// in-degree at dst (cnt), out-degree at src (deg)
__global__ void k_deg(const int* __restrict__ ei, int E,
                      float* __restrict__ cnt, float* __restrict__ deg) {
  int e = blockIdx.x * blockDim.x + threadIdx.x;
  if (e >= E) return;
  atomicAdd(&deg[ei[e]], 1.0f);
  atomicAdd(&cnt[ei[E + e]], 1.0f);
}

// deg -> dis = deg>0 ? rsqrt(max(deg,1)) : 0   (in place)
__global__ void k_dis(float* __restrict__ deg, int N) {
  int i = blockIdx.x * blockDim.x + threadIdx.x;
  if (i >= N) return;
  float d = deg[i];
  deg[i] = (d > 0.0f) ? rsqrtf(fmaxf(d, 1.0f)) : 0.0f;
}

// msum[dst,c] += x[src,c]   (thread = edge*96 + c, fully coalesced)
__global__ void k_sage_scatter(const int* __restrict__ ei, int E,
                               const float* __restrict__ x, float* __restrict__ msum) {
  int i = blockIdx.x * blockDim.x + threadIdx.x;
  if (i >= E * 96) return;
  int e = i / 96, c = i - e * 96;
  int s = ei[e], d = ei[E + e];
  atomicAdd(&msum[(long)d * 96 + c], x[(long)s * 96 + c]);
}

// out[dst,c] += norm(e) * in[src,c], norm = -(dis[src]*dis[dst])
__global__ void k_cheb_scatter(const int* __restrict__ ei, int E,
                               const float* __restrict__ dis,
                               const float* __restrict__ in, float* __restrict__ out) {
  int i = blockIdx.x * blockDim.x + threadIdx.x;
  if (i >= E * 96) return;
  int e = i / 96, c = i - e * 96;
  int s = ei[e], d = ei[E + e];
  float nrm = -(dis[s] * dis[d]);
  atomicAdd(&out[(long)d * 96 + c], nrm * in[(long)s * 96 + c]);
}

// per-edge logits + sortable-uint segment max at dst
__global__ void k_logits(const int* __restrict__ ei, int E,
                         const float* __restrict__ q, const float* __restrict__ k,
                         float* __restrict__ logits, unsigned* __restrict__ mbits) {
  int e = blockIdx.x * blockDim.x + threadIdx.x;
  if (e >= E) return;
  int s = ei[e], d = ei[E + e];
  const float4* qp = (const float4*)(q + (long)d * 64);
  const float4* kp = (const float4*)(k + (long)s * 64);
  float acc = 0.0f;
#pragma unroll
  for (int i = 0; i < 16; ++i) {
    float4 a = qp[i], b = kp[i];
    acc += a.x * b.x + a.y * b.y + a.z * b.z + a.w * b.w;
  }
  float lg = acc * 0.125f;  // 1/sqrt(64)
  logits[e] = lg;
  unsigned u = __float_as_uint(lg);
  u = (u & 0x80000000u) ? ~u : (u | 0x80000000u);  // order-preserving encode
  atomicMax(&mbits[d], u);
}

// e = exp(logit - m[dst]) ; ssum[dst] += e ; logits overwritten with e
__global__ void k_expsum(const int* __restrict__ ei, int E,
                         float* __restrict__ logits, const unsigned* __restrict__ mbits,
                         float* __restrict__ ssum) {
  int e = blockIdx.x * blockDim.x + threadIdx.x;
  if (e >= E) return;
  int d = ei[E + e];
  unsigned u = mbits[d];
  u = (u & 0x80000000u) ? (u & 0x7FFFFFFFu) : ~u;  // decode
  float m = __uint_as_float(u);
  if (!__builtin_isfinite(m)) m = 0.0f;
  float ev = expf(logits[e] - m);
  logits[e] = ev;
  atomicAdd(&ssum[d], ev);
}

// agg[dst,c] += alpha(e) * v[src,c]
__global__ void k_attn_scatter(const int* __restrict__ ei, int E,
                               const float* __restrict__ eweights,
                               const float* __restrict__ ssum,
                               const float* __restrict__ v, float* __restrict__ agg) {
  int i = blockIdx.x * blockDim.x + threadIdx.x;
  if (i >= E * 64) return;
  int e = i >> 6, c = i & 63;
  int s = ei[e], d = ei[E + e];
  float alpha = eweights[e] / (ssum[d] + 1e-16f);
  atomicAdd(&agg[(long)d * 64 + c], alpha * v[(long)s * 64 + c]);
}

// ----------------------------------------------- WMMA [N,96]@[96,64]+bias -> out
__global__ __launch_bounds__(32) void k_lin64(const float* __restrict__ A,
                                              const float* __restrict__ W,
                                              const float* __restrict__ bias,
                                              float* __restrict__ out, int N) {
  int lane = threadIdx.x;
  int m0 = blockIdx.x * 16;
  int mr = m0 + (lane & 15); if (mr >= N) mr = N - 1;
  int khalf = (lane >> 4) << 1;   // 0 | 2 : K sub-pair per lane half
  int nlo = lane & 15;
  v8f zero = {0, 0, 0, 0, 0, 0, 0, 0};
  v8f acc[4];
#pragma unroll
  for (int t = 0; t < 4; ++t) acc[t] = zero;
  for (int k0 = 0; k0 < 96; k0 += 4) {
    int kk = k0 + khalf;
    long xo = (long)mr * 96 + kk;
    v2f a; a.x = A[xo]; a.y = A[xo + 1];
#pragma unroll
    for (int t = 0; t < 4; ++t) {
      int n = t * 16 + nlo;
      v2f b; b.x = W[kk * 64 + n]; b.y = W[(kk + 1) * 64 + n];
      acc[t] = WMMA4(a, b, acc[t]);
    }
  }
  int r0 = m0 + ((lane >> 4) << 3);
#pragma unroll
  for (int t = 0; t < 4; ++t) {
    int n = t * 16 + nlo;
#pragma unroll
    for (int r = 0; r < 8; ++r) {
      int row = r0 + r;
      if (row < N) out[(long)row * 64 + n] = acc[t][r] + bias[n];
    }
  }
}

// --------------------------------- fused branch kernel: writes cat[N,192]
__global__ __launch_bounds__(32) void k_f1(
    const float* __restrict__ x,
    const float* __restrict__ msum, const float* __restrict__ cnt,
    const float* __restrict__ tx1, const float* __restrict__ tx2,
    const float* __restrict__ agg,
    const float* __restrict__ Wsl, const float* __restrict__ bsl,
    const float* __restrict__ Wsr,
    const float* __restrict__ Wch, const float* __restrict__ bch,
    const float* __restrict__ Wsk, const float* __restrict__ bsk,
    float* __restrict__ cat, int N) {
  int lane = threadIdx.x;
  int m0 = blockIdx.x * 16;
  int branch = blockIdx.y;
  int mr = m0 + (lane & 15); if (mr >= N) mr = N - 1;
  int khalf = (lane >> 4) << 1;
  int nlo = lane & 15;
  v8f zero = {0, 0, 0, 0, 0, 0, 0, 0};
  v8f acc[4];
#pragma unroll
  for (int t = 0; t < 4; ++t) acc[t] = zero;
  int r0 = m0 + ((lane >> 4) << 3);

  if (branch == 0) {  // SAGE: gelu(xmean@Wl + x@Wr + b)
    float inv = 1.0f / fmaxf(cnt[mr], 1.0f);
    for (int k0 = 0; k0 < 96; k0 += 4) {
      int kk = k0 + khalf;
      long xo = (long)mr * 96 + kk;
      v2f am; am.x = msum[xo] * inv; am.y = msum[xo + 1] * inv;
      v2f ax; ax.x = x[xo];          ax.y = x[xo + 1];
#pragma unroll
      for (int t = 0; t < 4; ++t) {
        int n = t * 16 + nlo;
        v2f bl; bl.x = Wsl[kk * 64 + n]; bl.y = Wsl[(kk + 1) * 64 + n];
        v2f br; br.x = Wsr[kk * 64 + n]; br.y = Wsr[(kk + 1) * 64 + n];
        acc[t] = WMMA4(am, bl, acc[t]);
        acc[t] = WMMA4(ax, br, acc[t]);
      }
    }
#pragma unroll
    for (int t = 0; t < 4; ++t) {
      int n = t * 16 + nlo;
#pragma unroll
      for (int r = 0; r < 8; ++r) {
        int row = r0 + r;
        if (row < N) cat[(long)row * 192 + n] = gelu_exact(acc[t][r] + bsl[n]);
      }
    }
  } else if (branch == 1) {  // Cheb K=3: gelu(x@W0 + Tx1@W1 + (2*tx2-x)@W2 + b)
    const float* W0 = Wch;
    const float* W1 = Wch + 96 * 64;
    const float* W2 = Wch + 2 * 96 * 64;
    for (int k0 = 0; k0 < 96; k0 += 4) {
      int kk = k0 + khalf;
      long xo = (long)mr * 96 + kk;
      float x0 = x[xo], x1 = x[xo + 1];
      v2f a0; a0.x = x0;                     a0.y = x1;
      v2f a1; a1.x = tx1[xo];                a1.y = tx1[xo + 1];
      v2f a2; a2.x = 2.0f * tx2[xo] - x0;    a2.y = 2.0f * tx2[xo + 1] - x1;
#pragma unroll
      for (int t = 0; t < 4; ++t) {
        int n = t * 16 + nlo;
        v2f b0; b0.x = W0[kk * 64 + n]; b0.y = W0[(kk + 1) * 64 + n];
        v2f b1; b1.x = W1[kk * 64 + n]; b1.y = W1[(kk + 1) * 64 + n];
        v2f b2; b2.x = W2[kk * 64 + n]; b2.y = W2[(kk + 1) * 64 + n];
        acc[t] = WMMA4(a0, b0, acc[t]);
        acc[t] = WMMA4(a1, b1, acc[t]);
        acc[t] = WMMA4(a2, b2, acc[t]);
      }
    }
#pragma unroll
    for (int t = 0; t < 4; ++t) {
      int n = t * 16 + nlo;
#pragma unroll
      for (int r = 0; r < 8; ++r) {
        int row = r0 + r;
        if (row < N) cat[(long)row * 192 + 64 + n] = gelu_exact(acc[t][r] + bch[n]);
      }
    }
  } else {  // attention: gelu(agg + x@Wskip + b)
    for (int k0 = 0; k0 < 96; k0 += 4) {
      int kk = k0 + khalf;
      long xo = (long)mr * 96 + kk;
      v2f a; a.x = x[xo]; a.y = x[xo + 1];
#pragma unroll
      for (int t = 0; t < 4; ++t) {
        int n = t * 16 + nlo;
        v2f b; b.x = Wsk[kk * 64 + n]; b.y = Wsk[(kk + 1) * 64 + n];
        acc[t] = WMMA4(a, b, acc[t]);
      }
    }
#pragma unroll
    for (int t = 0; t < 4; ++t) {
      int n = t * 16 + nlo;
#pragma unroll
      for (int r = 0; r < 8; ++r) {
        int row = r0 + r;
        if (row < N)
          cat[(long)row * 192 + 128 + n] =
              gelu_exact(acc[t][r] + agg[(long)row * 64 + n] + bsk[n]);
      }
    }
  }
}

// ---- fused gate + residual + LayerNorm + GELU: out = gelu(LN(sig(cat@Wg+bg)*cat + x@Wres+bres))
__global__ __launch_bounds__(32) void k_f2(
    const float* __restrict__ x, const float* __restrict__ cat,
    const float* __restrict__ Wg, const float* __restrict__ bg,
    const float* __restrict__ Wres, const float* __restrict__ bres,
    const float* __restrict__ lng, const float* __restrict__ lnb,
    float* __restrict__ out, int N) {
  __shared__ float so[16 * 192];
  __shared__ float smu[16], srs[16];
  int lane = threadIdx.x;
  int m0 = blockIdx.x * 16;
  int mr = m0 + (lane & 15); if (mr >= N) mr = N - 1;
  int khalf = (lane >> 4) << 1;
  int nlo = lane & 15;
  v8f zero = {0, 0, 0, 0, 0, 0, 0, 0};

  for (int t = 0; t < 12; ++t) {
    int n = t * 16 + nlo;
    v8f ag = zero, ar = zero;
    for (int k0 = 0; k0 < 192; k0 += 4) {  // cat @ Wg
      int kk = k0 + khalf;
      long co = (long)mr * 192 + kk;
      v2f a; a.x = cat[co]; a.y = cat[co + 1];
      v2f b; b.x = Wg[kk * 192 + n]; b.y = Wg[(kk + 1) * 192 + n];
      ag = WMMA4(a, b, ag);
    }
    for (int k0 = 0; k0 < 96; k0 += 4) {   // x @ Wres
      int kk = k0 + khalf;
      long xo = (long)mr * 96 + kk;
      v2f a; a.x = x[xo]; a.y = x[xo + 1];
      v2f b; b.x = Wres[kk * 192 + n]; b.y = Wres[(kk + 1) * 192 + n];
      ar = WMMA4(a, b, ar);
    }
    int r0 = (lane >> 4) << 3;
#pragma unroll
    for (int r = 0; r < 8; ++r) {
      int row = r0 + r;
      int grow = m0 + row;
      float cv = (grow < N) ? cat[(long)grow * 192 + n] : 0.0f;
      float gate = 1.0f / (1.0f + expf(-(ag[r] + bg[n])));
      so[row * 192 + n] = gate * cv + ar[r] + bres[n];
    }
  }
  __syncthreads();

  {  // per-row mean/var: lane pair (row, half) each sums 96 elements
    int row = lane & 15, half = lane >> 4;
    float s = 0.0f, ss = 0.0f;
    int base = row * 192 + half * 96;
    for (int c = 0; c < 96; ++c) { float v = so[base + c]; s += v; ss += v * v; }
    s += __shfl_xor(s, 16, 32);
    ss += __shfl_xor(ss, 16, 32);
    if (half == 0) {
      float mu = s * (1.0f / 192.0f);
      float var = ss * (1.0f / 192.0f) - mu * mu;
      smu[row] = mu;
      srs[row] = rsqrtf(var + 1e-5f);
    }
  }
  __syncthreads();

  for (int idx = lane; idx < 16 * 192; idx += 32) {
    int row = idx / 192, col = idx - row * 192;
    int grow = m0 + row;
    if (grow < N) {
      float v = (so[idx] - smu[row]) * srs[row] * lng[col] + lnb[col];
      out[(long)grow * 192 + col] = gelu_exact(v);
    }
  }
}

// ---------------------------------------------------------------- launcher
extern "C" void kernel_launch(void* const* d_in, const int* in_sizes, int n_in,
                              void* d_out, int out_size, void* d_ws, size_t ws_size,
                              hipStream_t stream) {
  const float* x    = (const float*)d_in[0];
  const int*   ei   = (const int*)  d_in[1];
  const float* Wsl  = (const float*)d_in[2];
  const float* bsl  = (const float*)d_in[3];
  const float* Wsr  = (const float*)d_in[4];
  const float* Wch  = (const float*)d_in[5];
  const float* bch  = (const float*)d_in[6];
  const float* Wq   = (const float*)d_in[7];
  const float* bq   = (const float*)d_in[8];
  const float* Wk   = (const float*)d_in[9];
  const float* bk   = (const float*)d_in[10];
  const float* Wv   = (const float*)d_in[11];
  const float* bv   = (const float*)d_in[12];
  const float* Wsk  = (const float*)d_in[13];
  const float* bsk  = (const float*)d_in[14];
  const float* Wg   = (const float*)d_in[15];
  const float* bg   = (const float*)d_in[16];
  const float* Wres = (const float*)d_in[17];
  const float* bres = (const float*)d_in[18];
  const float* lng  = (const float*)d_in[19];
  const float* lnb  = (const float*)d_in[20];

  const int N = in_sizes[0] / 96;
  const int E = in_sizes[1] / 2;

  float* ws = (float*)d_ws;
  size_t o = 0;
  float* msum = ws + o; o += (size_t)N * 96;
  float* tx1  = ws + o; o += (size_t)N * 96;
  float* tx2  = ws + o; o += (size_t)N * 96;
  float* qb   = ws + o; o += (size_t)N * 64;
  float* kb   = ws + o; o += (size_t)N * 64;
  float* vb   = ws + o; o += (size_t)N * 64;
  float* agg  = ws + o; o += (size_t)N * 64;
  float* cnt  = ws + o; o += (size_t)N;
  float* deg  = ws + o; o += (size_t)N;   // becomes dis in place
  unsigned* mbits = (unsigned*)(ws + o); o += (size_t)N;
  float* ssum = ws + o; o += (size_t)N;
  size_t zeroEnd = o;                      // everything before cat gets zeroed
  float* catb = ws + o; o += (size_t)N * 192;
  float* lg   = ws + o; o += (size_t)E;
  if (o * sizeof(float) > ws_size) return;  // workspace too small: bail safely

  const int NT = (N + 15) / 16;
  const unsigned gE   = (unsigned)((E + 255) / 256);
  const unsigned gN   = (unsigned)((N + 255) / 256);
  const unsigned gE96 = (unsigned)(((long)E * 96 + 255) / 256);
  const unsigned gE64 = (unsigned)(((long)E * 64 + 255) / 256);

  // 1) zero all accumulators (msum..ssum)
  k_zero<<<(unsigned)((zeroEnd + 255) / 256), 256, 0, stream>>>(ws, (long)zeroEnd);
  // 2) degrees, then dis
  k_deg<<<gE, 256, 0, stream>>>(ei, E, cnt, deg);
  k_dis<<<gN, 256, 0, stream>>>(deg, N);
  // 3) SAGE neighbor sum
  k_sage_scatter<<<gE96, 256, 0, stream>>>(ei, E, x, msum);
  // 4) q,k,v projections (WMMA)
  k_lin64<<<NT, 32, 0, stream>>>(x, Wq, bq, qb, N);
  k_lin64<<<NT, 32, 0, stream>>>(x, Wk, bk, kb, N);
  k_lin64<<<NT, 32, 0, stream>>>(x, Wv, bv, vb, N);
  // 5) Cheb pass 1: tx1 = segsum(norm * x[src])
  k_cheb_scatter<<<gE96, 256, 0, stream>>>(ei, E, deg, x, tx1);
  // 6) edge logits + segment max
  k_logits<<<gE, 256, 0, stream>>>(ei, E, qb, kb, lg, mbits);
  // 7) Cheb pass 2: tx2 = segsum(norm * tx1[src])
  k_cheb_scatter<<<gE96, 256, 0, stream>>>(ei, E, deg, tx1, tx2);
  // 8) exp + segment sum
  k_expsum<<<gE, 256, 0, stream>>>(ei, E, lg, mbits, ssum);
  // 9) attention aggregation
  k_attn_scatter<<<gE64, 256, 0, stream>>>(ei, E, lg, ssum, vb, agg);
  // 10) three branches -> cat (WMMA)
  k_f1<<<dim3(NT, 3), 32, 0, stream>>>(x, msum, cnt, tx1, tx2, agg,
                                       Wsl, bsl, Wsr, Wch, bch, Wsk, bsk, catb, N);
  // 11) gate + residual + LayerNorm + GELU (WMMA)
  k_f2<<<NT, 32, 0, stream>>>(x, catb, Wg, bg, Wres, bres, lng, lnb,
                              (float*)d_out, N);
}